// RGATLayer_11553462026815
// MI455X (gfx1250) — compile-verified
//
#include <hip/hip_runtime.h>
#include <hip/hip_bf16.h>

typedef __attribute__((ext_vector_type(2))) float v2f;
typedef __attribute__((ext_vector_type(4))) float v4f;
typedef __attribute__((ext_vector_type(8))) float v8f;

#define WMMA_F32X4(a, b, c) \
  __builtin_amdgcn_wmma_f32_16x16x4_f32(false, (a), false, (b), (short)0, (c), false, false)

constexpr int B_ = 4, N_ = 1024, D_ = 128;
constexpr float SLOPE = 0.2f;
constexpr float NEG_INF_ = -9.0e15f;

constexpr int LSTR = 132;          // padded LDS row stride (floats): bank-conflict free
constexpr int TILE_F = 16 * LSTR;  // floats per 16-row LDS tile

// ---------------------------------------------------------------------------
// Kernel 1: h = hidden @ W     [B,N,D] x [D,D] -> [B,N,D]
// One wave computes one 16x16 tile of h; 8 waves/block cover the D dimension.
// ---------------------------------------------------------------------------
__global__ __launch_bounds__(256) void rgat_hw_gemm(const float* __restrict__ hidden,
                                                    const float* __restrict__ W,
                                                    float* __restrict__ hout) {
  const int wave = threadIdx.x >> 5;
  const int lane = threadIdx.x & 31;
  const int m    = lane & 15;
  const int hl   = lane >> 4;
  const int b    = blockIdx.x >> 6;
  const int it   = blockIdx.x & 63;
  const int i0   = it * 16;
  const int c0   = wave * 16;

  const float* Arow = hidden + (size_t)(b * N_ + i0 + m) * D_ + 2 * hl;
  v8f acc = {};
#pragma unroll
  for (int k = 0; k < D_; k += 4) {
    v2f a = *(const v2f*)(Arow + k);
    v2f bv;
    bv.x = W[(size_t)(k + 2 * hl) * D_ + c0 + m];
    bv.y = W[(size_t)(k + 2 * hl + 1) * D_ + c0 + m];
    acc = WMMA_F32X4(a, bv, acc);
  }
  float* orow = hout + (size_t)(b * N_ + i0) * D_ + c0 + m;
#pragma unroll
  for (int v = 0; v < 8; ++v)
    orow[(size_t)(v + 8 * hl) * D_] = acc[v];
}

// ---------------------------------------------------------------------------
// Kernel 2: fused relation-scores + select + online-softmax + alpha@h.
// One wave per (batch, 16-row i-tile); streams 16-wide j-tiles.
// A-operands (h_i .* a_r) precomputed in LDS; H_j tiles streamed into LDS
// with async global->LDS DMA, double-buffered and pipelined one tile ahead.
// ---------------------------------------------------------------------------
__global__ __launch_bounds__(32) void rgat_attn(const float* __restrict__ h,
                                                const int* __restrict__ adj,
                                                const float* __restrict__ a_params,
                                                const float* __restrict__ amplifier,
                                                float* __restrict__ out) {
  __shared__ float sA[4 * TILE_F];   // Atilde_r: 4 x 16 x 128 (padded)
  __shared__ float sH[2 * TILE_F];   // double-buffered H_j tile
  __shared__ float sP[16][17];       // P tile staging (C-layout -> A-layout)

  const int lane = threadIdx.x & 31;
  const int m    = lane & 15;
  const int hl   = lane >> 4;
  const int b    = blockIdx.x >> 6;
  const int it   = blockIdx.x & 63;
  const int i0   = it * 16;

  const float amp0 = amplifier[0], amp1 = amplifier[1];
  const float amp2 = amplifier[2], amp3 = amplifier[3];
  const float* hb   = h + (size_t)b * N_ * D_;
  const int*   adjb = adj + (size_t)b * N_ * N_;
  const unsigned sHbase = (unsigned)(size_t)sH;   // LDS byte offset of sH

  // ---- prologue: Atilde_r = h_i .* a_r into LDS (coalesced float4) --------
#pragma unroll
  for (int t = 0; t < 64; ++t) {
    const int idx = t * 32 + lane;           // 0..2047 float4 units
    const int r   = idx >> 9;
    const int row = (idx >> 5) & 15;
    const int c4  = idx & 31;
    v4f hv = *(const v4f*)(hb + (size_t)(i0 + row) * D_ + c4 * 4);
    v4f av = *(const v4f*)(a_params + r * D_ + c4 * 4);
    *(v4f*)(sA + (r * 16 + row) * LSTR + c4 * 4) = hv * av;
  }
  __syncthreads();

  float mrow[8], lrow[8];
  v8f O[8];
#pragma unroll
  for (int v = 0; v < 8; ++v) { mrow[v] = -3.0e38f; lrow[v] = 0.f; O[v] = (v8f){}; }

  const float* sA0 = sA + 0 * TILE_F + m * LSTR + 2 * hl;
  const float* sA1 = sA + 1 * TILE_F + m * LSTR + 2 * hl;
  const float* sA2 = sA + 2 * TILE_F + m * LSTR + 2 * hl;
  const float* sA3 = sA + 3 * TILE_F + m * LSTR + 2 * hl;

  // async-stage one 16x128 H_j tile into LDS buffer `buf` (16 x b128 / lane)
  auto stage = [&](int jt, int buf) {
    const float* gsrc = hb + (size_t)(jt * 16) * D_;
#pragma unroll
    for (int t = 0; t < 16; ++t) {
      const int idx = t * 32 + lane;         // 0..511 float4 units
      const int row = idx >> 5;
      const int c4  = idx & 31;
      const unsigned ldsoff = sHbase + (unsigned)(buf * TILE_F + row * LSTR + c4 * 4) * 4u;
      const float*   g      = gsrc + (size_t)row * D_ + c4 * 4;
      asm volatile("global_load_async_to_lds_b128 %0, %1, off"
                   :: "v"(ldsoff), "v"(g) : "memory");
    }
  };

  stage(0, 0);

  for (int jt = 0; jt < N_ / 16; ++jt) {
    const int cur = jt & 1;
    const int j0  = jt * 16;

    if (jt + 1 < N_ / 16) {
      __builtin_prefetch((const void*)(adjb + (size_t)(i0 + m) * N_ + j0 + 16), 0, 1);
      stage(jt + 1, cur ^ 1);
      asm volatile("s_wait_asynccnt 0x10" ::: "memory");  // current tile landed
    } else {
      asm volatile("s_wait_asynccnt 0x0" ::: "memory");
    }

    // ---- scores for all 4 relations: S_r = Atilde_r @ Hj^T (all from LDS) --
    v8f S0 = {}, S1 = {}, S2 = {}, S3 = {};
    const float* sHb = sH + cur * TILE_F + m * LSTR + 2 * hl;
#pragma unroll
    for (int k = 0; k < D_; k += 4) {
      v2f bv = *(const v2f*)(sHb + k);
      S0 = WMMA_F32X4(*(const v2f*)(sA0 + k), bv, S0);
      S1 = WMMA_F32X4(*(const v2f*)(sA1 + k), bv, S1);
      S2 = WMMA_F32X4(*(const v2f*)(sA2 + k), bv, S2);
      S3 = WMMA_F32X4(*(const v2f*)(sA3 + k), bv, S3);
    }

    // ---- per-element relation select + leaky relu + amplifier + mask ------
    float p[8];
#pragma unroll
    for (int v = 0; v < 8; ++v) {
      const int irow = i0 + v + 8 * hl;
      const int av   = adjb[(size_t)irow * N_ + j0 + m];
      const int r    = min(max(av - 1, 0), 3);
      float e = (r == 0) ? S0[v] : (r == 1) ? S1[v] : (r == 2) ? S2[v] : S3[v];
      e = (e >= 0.f) ? e : SLOPE * e;
      const float amp = (r == 0) ? amp0 : (r == 1) ? amp1 : (r == 2) ? amp2 : amp3;
      p[v] = (av > 0) ? e * amp : NEG_INF_;
    }

    // ---- online softmax over j (rows live across 16-lane N-groups) --------
#pragma unroll
    for (int v = 0; v < 8; ++v) {
      float t = p[v];
      t = fmaxf(t, __shfl_xor(t, 1, 32));
      t = fmaxf(t, __shfl_xor(t, 2, 32));
      t = fmaxf(t, __shfl_xor(t, 4, 32));
      t = fmaxf(t, __shfl_xor(t, 8, 32));
      const float mnew  = fmaxf(mrow[v], t);
      const float scale = __expf(mrow[v] - mnew);
      mrow[v] = mnew;
      const float pe = __expf(p[v] - mnew);
      p[v] = pe;
      float s = pe;
      s += __shfl_xor(s, 1, 32);
      s += __shfl_xor(s, 2, 32);
      s += __shfl_xor(s, 4, 32);
      s += __shfl_xor(s, 8, 32);
      lrow[v] = lrow[v] * scale + s;
#pragma unroll
      for (int nt = 0; nt < 8; ++nt) O[nt][v] *= scale;
    }

    // ---- transpose P (C-layout) into A-layout via LDS ---------------------
    __syncthreads();
#pragma unroll
    for (int v = 0; v < 8; ++v) sP[v + 8 * hl][m] = p[v];
    __syncthreads();

    // ---- O += P @ Hj  (M=i, K=j-local 16, N=d 128), Hj from LDS -----------
    const float* sHrow = sH + cur * TILE_F;
#pragma unroll
    for (int kk = 0; kk < 16; kk += 4) {
      v2f pa;
      pa.x = sP[m][kk + 2 * hl];
      pa.y = sP[m][kk + 2 * hl + 1];
      const float* r0 = sHrow + (kk + 2 * hl) * LSTR + m;
      const float* r1 = r0 + LSTR;
#pragma unroll
      for (int nt = 0; nt < 8; ++nt) {
        v2f bv;
        bv.x = r0[nt * 16];
        bv.y = r1[nt * 16];
        O[nt] = WMMA_F32X4(pa, bv, O[nt]);
      }
    }
  }

  // ---- normalize and store -----------------------------------------------
  float* orow = out + (size_t)(b * N_ + i0) * D_ + m;
#pragma unroll
  for (int v = 0; v < 8; ++v) {
    const float inv = 1.0f / lrow[v];
#pragma unroll
    for (int nt = 0; nt < 8; ++nt)
      orow[(size_t)(v + 8 * hl) * D_ + nt * 16] = O[nt][v] * inv;
  }
}

// ---------------------------------------------------------------------------
extern "C" void kernel_launch(void* const* d_in, const int* in_sizes, int n_in,
                              void* d_out, int out_size, void* d_ws, size_t ws_size,
                              hipStream_t stream) {
  const float* hidden    = (const float*)d_in[0];
  const int*   adj       = (const int*)d_in[1];
  // d_in[2] node_type_mask: unused by the reference computation
  const float* W         = (const float*)d_in[3];
  const float* a_params  = (const float*)d_in[4];
  const float* amplifier = (const float*)d_in[5];
  float* out = (float*)d_out;
  float* hW  = (float*)d_ws;                 // B*N*D floats = 2 MB scratch

  rgat_hw_gemm<<<dim3(B_ * (N_ / 16)), dim3(256), 0, stream>>>(hidden, W, hW);
  rgat_attn<<<dim3(B_ * (N_ / 16)), dim3(32), 0, stream>>>(hW, adj, a_params, amplifier, out);
}